// HarrisNet_54391465837293
// MI455X (gfx1250) — compile-verified
//
#include <hip/hip_runtime.h>
#include <math.h>

typedef float v2f __attribute__((ext_vector_type(2)));
typedef float v8f __attribute__((ext_vector_type(8)));

#define IMG 4096
#define TS  32           // output tile per workgroup (kernel 1)
#define IN_R 40          // input tile rows/cols (TS + 2*4 halo)
#define IN_S 41          // LDS stride (odd -> conflict-free row access)
#define P_R  48          // product plane rows (3 WMMA row blocks)
#define P_C  40          // product plane cols actually written
#define P_S  41
#define H_R  48          // horizontally-blurred plane rows
#define H_S  33

// ctrl block indices (u32 words), placed after R in workspace
#define C_HIST   0       // [0..255]
#define C_PREFIX 256
#define C_RANK   257
#define C_KEY0   258     // key for rank N/2-1
#define C_KEY1   259     // key for rank N/2
#define C_MED    260     // median as float bits

__device__ __forceinline__ unsigned f2key(float f) {
  unsigned b = __float_as_uint(f);
  return (b & 0x80000000u) ? ~b : (b | 0x80000000u);  // order-preserving
}
__device__ __forceinline__ float key2f(unsigned k) {
  unsigned b = (k & 0x80000000u) ? (k ^ 0x80000000u) : ~k;
  return __uint_as_float(b);
}

// ---------------------------------------------------------------------------
// Kernel 1: fused Sobel -> products -> separable 7x7 Gaussian (WMMA banded
// matmul, V_WMMA_F32_16X16X4_F32) -> Harris response R.
// One 32x32 output tile per 128-thread (4-wave) workgroup.
// ---------------------------------------------------------------------------
__global__ __launch_bounds__(128) void k_harris_R(const float* __restrict__ x,
                                                  float* __restrict__ Rb) {
  __shared__ float s_in[IN_R * IN_S];        //  6.6 KB
  __shared__ float s_p[3][P_R * P_S];        // 23.6 KB  Ix2,Iy2,IxIy (+zero pad)
  __shared__ float s_h[3][H_R * H_S];        // 19.0 KB  horizontally blurred

  const int tid  = threadIdx.x;
  const int lane = tid & 31;
  const int wid  = tid >> 5;
  const int lh2  = (lane >> 4) << 1;   // 0 (lanes 0-15) or 2 (lanes 16-31)
  const int lm   = lane & 15;
  const int bx = blockIdx.x * TS, by = blockIdx.y * TS;

  // normalized 1D gaussian, sigma=5, ksize=7  (separable: g2 = outer(g1,g1)/sum)
  float g[7]; float gs = 0.f;
#pragma unroll
  for (int i = 0; i < 7; ++i) { float d = (float)(i - 3); g[i] = __expf(-d * d * 0.02f); gs += g[i]; }
  float ginv = 1.f / gs;
#pragma unroll
  for (int i = 0; i < 7; ++i) g[i] *= ginv;

  // Lane-mapped banded weight matrix, 4-K slices (K padded 22->24).
  // Serves as B in the horizontal pass (B[j][n] = w[j-n]) and as A in the
  // vertical pass (A[m][j] = w[j-m]) thanks to the symmetric lane layouts.
  v2f wv[6];
#pragma unroll
  for (int ks = 0; ks < 6; ++ks) {
    int j = 4 * ks + lh2;
    int d0 = j - lm, d1 = j + 1 - lm;
    wv[ks].x = (d0 >= 0 && d0 < 7) ? g[d0] : 0.f;
    wv[ks].y = (d1 >= 0 && d1 < 7) ? g[d1] : 0.f;
  }

  // phase 0: load 40x40 input tile (halo 4), zero-padded (matches conv padding)
  for (int idx = tid; idx < IN_R * IN_R; idx += 128) {
    int r = idx / IN_R, c = idx % IN_R;
    int gy = by - 4 + r, gx = bx - 4 + c;
    float v = 0.f;
    if ((unsigned)gy < IMG && (unsigned)gx < IMG) v = x[(size_t)gy * IMG + gx];
    s_in[r * IN_S + c] = v;
  }
  __syncthreads();

  // phase 1: Sobel + products over the 38x38 region (halo 3 for gaussian),
  // zero-fill the WMMA padding so zero-weight lanes never see NaN garbage.
  for (int idx = tid; idx < P_R * P_C; idx += 128) {
    int r = idx / P_C, c = idx % P_C;
    float p0 = 0.f, p1 = 0.f, p2 = 0.f;
    if (r < 38 && c < 38) {
      const float* q = &s_in[r * IN_S + c];
      float a00 = q[0],        a01 = q[1],            a02 = q[2];
      float a10 = q[IN_S],                            a12 = q[IN_S + 2];
      float a20 = q[2 * IN_S], a21 = q[2 * IN_S + 1], a22 = q[2 * IN_S + 2];
      float Ix = (a02 - a00) + 2.f * (a12 - a10) + (a22 - a20);
      float Iy = (a20 - a00) + 2.f * (a21 - a01) + (a22 - a02);
      p0 = Ix * Ix; p1 = Iy * Iy; p2 = Ix * Iy;
    }
    s_p[0][r * P_S + c] = p0;
    s_p[1][r * P_S + c] = p1;
    s_p[2][r * P_S + c] = p2;
  }
  __syncthreads();

  // phase 2: horizontal 7-tap as banded matmul  h = P(16x24) x W(24x16)
  // 6 tiles (3 row blocks x 2 col blocks), round-robin over 4 waves.
  for (int t = wid; t < 6; t += 4) {
    int r0 = (t >> 1) * 16, n0 = (t & 1) * 16;
    for (int ch = 0; ch < 3; ++ch) {
      v8f acc = {};
#pragma unroll
      for (int ks = 0; ks < 6; ++ks) {
        // A 16x4 f32 layout: lane<16 holds K={4ks,4ks+1}, lane>=16 K={4ks+2,4ks+3}
        const float* q = &s_p[ch][(r0 + lm) * P_S + n0 + 4 * ks + lh2];
        v2f a; a.x = q[0]; a.y = q[1];
        acc = __builtin_amdgcn_wmma_f32_16x16x4_f32(
            false, a, false, wv[ks], (short)0, acc, false, false);
      }
      // C/D layout: VGPR v -> row r0+v (lanes 0-15) / r0+v+8 (lanes 16-31)
      float* hb = &s_h[ch][0];
#pragma unroll
      for (int v = 0; v < 8; ++v)
        hb[(r0 + v + (lh2 << 2)) * H_S + n0 + lm] = acc[v];
    }
  }
  __syncthreads();

  // phase 3: vertical 7-tap as banded matmul  S = W(16x24) x H(24x16), then R
  for (int t = wid; t < 4; t += 4) {
    int m0 = (t >> 1) * 16, n0 = (t & 1) * 16;
    v8f sxx = {}, syy = {}, sxy = {};
#pragma unroll
    for (int ks = 0; ks < 6; ++ks) {
      int jr = m0 + 4 * ks + lh2;  // B 4x16: lane<16 rows {j,j+1}, lane>=16 rows {j+2,j+3}
      const float* h0 = &s_h[0][jr * H_S + n0 + lm];
      const float* h1 = &s_h[1][jr * H_S + n0 + lm];
      const float* h2 = &s_h[2][jr * H_S + n0 + lm];
      v2f b0, b1, b2;
      b0.x = h0[0]; b0.y = h0[H_S];
      b1.x = h1[0]; b1.y = h1[H_S];
      b2.x = h2[0]; b2.y = h2[H_S];
      sxx = __builtin_amdgcn_wmma_f32_16x16x4_f32(false, wv[ks], false, b0, (short)0, sxx, false, false);
      syy = __builtin_amdgcn_wmma_f32_16x16x4_f32(false, wv[ks], false, b1, (short)0, syy, false, false);
      sxy = __builtin_amdgcn_wmma_f32_16x16x4_f32(false, wv[ks], false, b2, (short)0, sxy, false, false);
    }
#pragma unroll
    for (int v = 0; v < 8; ++v) {
      float Sxx = sxx[v], Syy = syy[v], Sxy = sxy[v];
      float tr = Sxx + Syy;
      float Rv = Sxx * Syy - Sxy * Sxy - 0.05f * tr * tr;
      int row = m0 + v + (lh2 << 2);
      Rb[(size_t)(by + row) * IMG + bx + n0 + lm] = Rv;
    }
  }
}

// ---------------------------------------------------------------------------
// Exact median via 8-bit radix select (4 passes), run for both middle ranks.
// ---------------------------------------------------------------------------
__global__ __launch_bounds__(256) void k_init(unsigned* __restrict__ ctrl, int sel) {
  int t = threadIdx.x;
  ctrl[C_HIST + t] = 0u;
  if (t == 0) {
    ctrl[C_PREFIX] = 0u;
    ctrl[C_RANK]   = sel ? 8388608u : 8388607u;  // N/2 and N/2-1
  }
}

__global__ __launch_bounds__(256) void k_hist(const float* __restrict__ R,
                                              unsigned* __restrict__ ctrl, int level) {
  __shared__ unsigned lh[256];
  lh[threadIdx.x] = 0u;
  __syncthreads();
  const unsigned prefix = ctrl[C_PREFIX];
  const int shift = 24 - 8 * level;
  const unsigned pshift = (unsigned)(shift + 8);
  const size_t N4 = (size_t)IMG * IMG / 4;
  const float4* __restrict__ R4 = (const float4*)R;
  const size_t stride = (size_t)gridDim.x * blockDim.x;
  for (size_t i = (size_t)blockIdx.x * blockDim.x + threadIdx.x; i < N4; i += stride) {
    if (i + stride < N4) __builtin_prefetch(&R4[i + stride], 0, 0);  // global_prefetch_b8
    float4 v = R4[i];
    float vals[4] = {v.x, v.y, v.z, v.w};
#pragma unroll
    for (int j = 0; j < 4; ++j) {
      unsigned key = f2key(vals[j]);
      bool ok = (level == 0) || ((key >> pshift) == prefix);
      if (ok) atomicAdd(&lh[(key >> shift) & 255u], 1u);
    }
  }
  __syncthreads();
  if (lh[threadIdx.x]) atomicAdd(&ctrl[C_HIST + threadIdx.x], lh[threadIdx.x]);
}

__global__ __launch_bounds__(256) void k_scan(unsigned* __restrict__ ctrl, int level, int sel) {
  __shared__ unsigned h[256], c[256];
  int t = threadIdx.x;
  unsigned hv = ctrl[C_HIST + t];
  h[t] = hv; c[t] = hv;
  __syncthreads();
  for (int off = 1; off < 256; off <<= 1) {
    unsigned add = (t >= off) ? c[t - off] : 0u;
    __syncthreads();
    c[t] += add;
    __syncthreads();
  }
  unsigned rank = ctrl[C_RANK];
  unsigned incl = c[t], excl = incl - h[t];
  if (rank >= excl && rank < incl) {          // unique winner digit
    ctrl[C_RANK] = rank - excl;
    unsigned np = (ctrl[C_PREFIX] << 8) | (unsigned)t;
    ctrl[C_PREFIX] = np;
    if (level == 3) ctrl[C_KEY0 + sel] = np;  // full 32-bit key recovered
  }
  __syncthreads();
  ctrl[C_HIST + t] = 0u;                      // ready for next level
}

__global__ void k_median(unsigned* __restrict__ ctrl) {
  if (threadIdx.x == 0) {
    float a = key2f(ctrl[C_KEY0]);
    float b = key2f(ctrl[C_KEY1]);
    ctrl[C_MED] = __float_as_uint(0.5f * (a + b));
  }
}

// ---------------------------------------------------------------------------
// Kernel 3: threshold + separable 7x7 max-pool + NMS. 64x64 tile / 256 thr.
// ---------------------------------------------------------------------------
__global__ __launch_bounds__(256) void k_nms(const float* __restrict__ R,
                                             const unsigned* __restrict__ ctrl,
                                             float* __restrict__ out) {
  __shared__ float rt[70 * 73];
  __shared__ float hm[70 * 65];
  const float med = __uint_as_float(ctrl[C_MED]);
  const int tid = threadIdx.x;
  const int bx = blockIdx.x * 64, by = blockIdx.y * 64;

  for (int idx = tid; idx < 70 * 70; idx += 256) {
    int r = idx / 70, c = idx % 70;
    int gy = by + r - 3, gx = bx + c - 3;
    float v;
    if ((unsigned)gy < IMG && (unsigned)gx < IMG) {
      float xv = R[(size_t)gy * IMG + gx];
      v = (xv >= med) ? xv : 0.f;             // Rt
    } else {
      v = -INFINITY;                          // reduce_window pad value
    }
    rt[r * 73 + c] = v;
  }
  __syncthreads();

  for (int idx = tid; idx < 70 * 64; idx += 256) {
    int r = idx / 64, n = idx % 64;
    const float* q = &rt[r * 73 + n];
    float m = q[0];
#pragma unroll
    for (int k = 1; k < 7; ++k) m = fmaxf(m, q[k]);
    hm[r * 65 + n] = m;
  }
  __syncthreads();

  for (int idx = tid; idx < 64 * 64; idx += 256) {
    int m_ = idx / 64, n = idx % 64;
    const float* q = &hm[m_ * 65 + n];
    float mp = q[0];
#pragma unroll
    for (int k = 1; k < 7; ++k) mp = fmaxf(mp, q[k * 65]);
    float rc = rt[(m_ + 3) * 73 + n + 3];
    out[(size_t)(by + m_) * IMG + bx + n] = (rc == mp) ? mp : 0.f;
  }
}

// ---------------------------------------------------------------------------
extern "C" void kernel_launch(void* const* d_in, const int* in_sizes, int n_in,
                              void* d_out, int out_size, void* d_ws, size_t ws_size,
                              hipStream_t stream) {
  (void)in_sizes; (void)n_in; (void)out_size; (void)ws_size;
  const float* x = (const float*)d_in[0];                 // 1x1x4096x4096 f32
  float* Rb = (float*)d_ws;                               // 64 MiB response map
  unsigned* ctrl = (unsigned*)((char*)d_ws + (size_t)IMG * IMG * sizeof(float));
  float* out = (float*)d_out;

  dim3 g1(IMG / TS, IMG / TS);
  k_harris_R<<<g1, 128, 0, stream>>>(x, Rb);

  for (int sel = 0; sel < 2; ++sel) {
    k_init<<<1, 256, 0, stream>>>(ctrl, sel);
    for (int lv = 0; lv < 4; ++lv) {
      k_hist<<<1024, 256, 0, stream>>>(Rb, ctrl, lv);
      k_scan<<<1, 256, 0, stream>>>(ctrl, lv, sel);
    }
  }
  k_median<<<1, 32, 0, stream>>>(ctrl);

  dim3 g3(IMG / 64, IMG / 64);
  k_nms<<<g3, 256, 0, stream>>>(Rb, ctrl, out);
}